// RecurrentGraphChef_22179211117286
// MI455X (gfx1250) — compile-verified
//
#include <hip/hip_runtime.h>
#include <hip/hip_bf16.h>

// ---------------------------------------------------------------------------
// RecurrentGraphChef on MI455X (gfx1250, wave32, WMMA).
//
// Forward-value semantics of hard_softmax(+straight-through) == one-hot of
// argmax, so node state is an integer in [0,64). Per iteration:
//   1) 64-bin histogram of in-neighbor states (u32 atomics, L2-resident)
//   2) z = min(cnt,10) @ W_rec_top  (WMMA f32 16x16x4)  + W_rec_bot[state]
//   3) state = argmax BN(z)
// Output = 64x8 precomputed lookup table indexed by final state.
// ---------------------------------------------------------------------------

#define NN   100000
#define EE   3200000
#define CIN  128
#define SS   64
#define HH   16
#define OOUT 8
#define ITERS 12
#define BN_EPS 1e-5f
#define B_CLAMP 10.0f

typedef float v2f __attribute__((ext_vector_type(2)));
typedef float v8f __attribute__((ext_vector_type(8)));

// ---------------------------------------------------------------------------
// Init: state[n] = argmax_j ( x[n,:] @ W_in[:,j] + b_in[j] ),  K = 128
// One wave -> 16 nodes x 64 logits via v_wmma_f32_16x16x4_f32.
// ---------------------------------------------------------------------------
__global__ __launch_bounds__(256)
void init_state_kernel(const float* __restrict__ x,
                       const float* __restrict__ W_in,
                       const float* __restrict__ b_in,
                       int* __restrict__ state, int n)
{
    __shared__ float lds[8][16][65];              // +1 pad: kill bank conflicts
    const int lane = threadIdx.x & 31;
    const int wave = threadIdx.x >> 5;
    const int m    = lane & 15;                   // row within tile / col within tile
    const int kh   = lane >> 4;                   // K-half selector (0 or 1)
    const int tile = blockIdx.x * 8 + wave;
    const int node = tile * 16 + m;
    const int nodec = node < n ? node : n - 1;    // clamp addr; stores are guarded
    const float* xrow = x + (size_t)nodec * CIN;

    for (int cn = 0; cn < 4; ++cn) {              // 4 column tiles of 16
        v8f acc = {0.f,0.f,0.f,0.f,0.f,0.f,0.f,0.f};
        #pragma unroll 8
        for (int k0 = 0; k0 < CIN; k0 += 4) {
            const int k = k0 + kh * 2;
            v2f a; a.x = xrow[k];                  a.y = xrow[k + 1];
            v2f b; b.x = W_in[k * SS + cn*16 + m]; b.y = W_in[(k+1) * SS + cn*16 + m];
            acc = __builtin_amdgcn_wmma_f32_16x16x4_f32(
                      false, a, false, b, (short)0, acc, false, false);
        }
        #pragma unroll
        for (int r = 0; r < 8; ++r)
            lds[wave][r + 8*kh][cn*16 + m] = acc[r];
    }
    __syncthreads();
    if (lane < 16 && node < n) {
        float best = -3.4e38f; int bi = 0;
        #pragma unroll 8
        for (int j = 0; j < SS; ++j) {
            float z = lds[wave][m][j] + b_in[j];
            if (z > best) { best = z; bi = j; }   // strict '>' => first max (jnp.argmax)
        }
        state[node] = bi;
    }
}

// ---------------------------------------------------------------------------
// Histogram: cnt[dst][state[src]] += 1 over 3.2M edges. Count array (25.6MB)
// fits in the 192MB L2 -> atomics resolve on-chip.
// edge_index layout: [src_0..src_{E-1}, dst_0..dst_{E-1}] (int32 per harness).
// ---------------------------------------------------------------------------
__global__ __launch_bounds__(256)
void edge_hist_kernel(const int* __restrict__ ei,
                      const int* __restrict__ state,
                      unsigned* __restrict__ cnt, int e)
{
    int t = blockIdx.x * blockDim.x + threadIdx.x;
    if (t < e) {
        int s = ei[t];
        int d = ei[e + t];
        int st = state[s];                         // 400KB state array: L2 hot
        atomicAdd(&cnt[(size_t)d * SS + st], 1u);
    }
}

// ---------------------------------------------------------------------------
// Recurrent update: z = min(cnt,10) @ W_rec[0:64,:]  (WMMA, K=64)
//                   + W_rec[64+old_state, :] + b_rec ;  state = argmax BN(z)
// ---------------------------------------------------------------------------
__global__ __launch_bounds__(256)
void update_state_kernel(const unsigned* __restrict__ cnt,
                         const float* __restrict__ W_rec,
                         const float* __restrict__ b_rec,
                         const float* __restrict__ bn_g,
                         const float* __restrict__ bn_b,
                         const float* __restrict__ bn_m,
                         const float* __restrict__ bn_v,
                         int* __restrict__ state, int n)
{
    __shared__ float lds[8][16][65];
    const int lane = threadIdx.x & 31;
    const int wave = threadIdx.x >> 5;
    const int m    = lane & 15;
    const int kh   = lane >> 4;
    const int tile = blockIdx.x * 8 + wave;
    const int node = tile * 16 + m;
    const int nodec = node < n ? node : n - 1;
    const unsigned* crow = cnt + (size_t)nodec * SS;

    for (int cn = 0; cn < 4; ++cn) {
        v8f acc = {0.f,0.f,0.f,0.f,0.f,0.f,0.f,0.f};
        #pragma unroll
        for (int k0 = 0; k0 < SS; k0 += 4) {
            const int k = k0 + kh * 2;
            v2f a; a.x = fminf((float)crow[k],   B_CLAMP);
                   a.y = fminf((float)crow[k+1], B_CLAMP);
            v2f b; b.x = W_rec[k * SS + cn*16 + m];
                   b.y = W_rec[(k+1) * SS + cn*16 + m];
            acc = __builtin_amdgcn_wmma_f32_16x16x4_f32(
                      false, a, false, b, (short)0, acc, false, false);
        }
        #pragma unroll
        for (int r = 0; r < 8; ++r)
            lds[wave][r + 8*kh][cn*16 + m] = acc[r];
    }
    __syncthreads();
    if (lane < 16 && node < n) {
        const int os = state[node];                       // read own old state
        const float* wb = W_rec + (size_t)(SS + os) * SS; // W_rec_bot row
        float best = -3.4e38f; int bi = 0;
        #pragma unroll 8
        for (int j = 0; j < SS; ++j) {
            float z  = lds[wave][m][j] + b_rec[j] + wb[j];
            float bn = (z - bn_m[j]) * rsqrtf(bn_v[j] + BN_EPS) * bn_g[j] + bn_b[j];
            if (bn > best) { best = bn; bi = j; }
        }
        state[node] = bi;                                 // write after own read only
    }
}

// ---------------------------------------------------------------------------
// Output table: table[k] = relu(BN(Wo1[k,:]+bo1)) @ Wo2 + bo2   (64 x 8)
// ---------------------------------------------------------------------------
__global__ void build_table_kernel(const float* __restrict__ Wo1,
                                   const float* __restrict__ bo1,
                                   const float* __restrict__ bno_g,
                                   const float* __restrict__ bno_b,
                                   const float* __restrict__ bno_m,
                                   const float* __restrict__ bno_v,
                                   const float* __restrict__ Wo2,
                                   const float* __restrict__ bo2,
                                   float* __restrict__ table)
{
    int k = threadIdx.x;
    if (k < SS) {
        float h[HH];
        #pragma unroll
        for (int i = 0; i < HH; ++i) {
            float v = Wo1[k * HH + i] + bo1[i];
            v = (v - bno_m[i]) * rsqrtf(bno_v[i] + BN_EPS) * bno_g[i] + bno_b[i];
            h[i] = fmaxf(v, 0.f);
        }
        #pragma unroll
        for (int o = 0; o < OOUT; ++o) {
            float acc = bo2[o];
            #pragma unroll
            for (int i = 0; i < HH; ++i) acc += h[i] * Wo2[i * OOUT + o];
            table[k * OOUT + o] = acc;
        }
    }
}

__global__ __launch_bounds__(256)
void gather_out_kernel(const int* __restrict__ state,
                       const float* __restrict__ table,
                       float* __restrict__ out, int total)
{
    int t = blockIdx.x * blockDim.x + threadIdx.x;
    if (t < total) {
        int n = t >> 3, o = t & 7;
        out[t] = table[state[n] * OOUT + o];
    }
}

// ---------------------------------------------------------------------------
extern "C" void kernel_launch(void* const* d_in, const int* in_sizes, int n_in,
                              void* d_out, int out_size, void* d_ws, size_t ws_size,
                              hipStream_t stream)
{
    const float* x     = (const float*)d_in[0];
    const int*   ei    = (const int*)  d_in[1];   // edge_index, flat [2,E]
    const float* W_in  = (const float*)d_in[2];
    const float* b_in  = (const float*)d_in[3];
    const float* W_rec = (const float*)d_in[4];
    const float* b_rec = (const float*)d_in[5];
    const float* bn_g  = (const float*)d_in[6];
    const float* bn_b  = (const float*)d_in[7];
    const float* bn_m  = (const float*)d_in[8];
    const float* bn_v  = (const float*)d_in[9];
    const float* Wo1   = (const float*)d_in[10];
    const float* bo1   = (const float*)d_in[11];
    const float* bno_g = (const float*)d_in[12];
    const float* bno_b = (const float*)d_in[13];
    const float* bno_m = (const float*)d_in[14];
    const float* bno_v = (const float*)d_in[15];
    const float* Wo2   = (const float*)d_in[16];
    const float* bo2   = (const float*)d_in[17];
    float* out = (float*)d_out;

    // workspace: counts [N*64 u32] | state [N i32] | table [64*8 f32]
    unsigned* cnt   = (unsigned*)d_ws;
    int*      state = (int*)((char*)d_ws + (size_t)NN * SS * sizeof(unsigned));
    float*    table = (float*)((char*)state + (size_t)NN * sizeof(int));

    const int tiles      = (NN + 15) / 16;
    const int gemm_blocks = (tiles + 7) / 8;                 // 8 waves/block
    const int edge_blocks = (EE + 255) / 256;

    init_state_kernel<<<gemm_blocks, 256, 0, stream>>>(x, W_in, b_in, state, NN);

    for (int it = 0; it < ITERS; ++it) {
        hipMemsetAsync(cnt, 0, (size_t)NN * SS * sizeof(unsigned), stream);
        edge_hist_kernel<<<edge_blocks, 256, 0, stream>>>(ei, state, cnt, EE);
        update_state_kernel<<<gemm_blocks, 256, 0, stream>>>(
            cnt, W_rec, b_rec, bn_g, bn_b, bn_m, bn_v, state, NN);
    }

    build_table_kernel<<<1, 64, 0, stream>>>(Wo1, bo1, bno_g, bno_b, bno_m, bno_v,
                                             Wo2, bo2, table);
    gather_out_kernel<<<(NN * OOUT + 255) / 256, 256, 0, stream>>>(
        state, table, out, NN * OOUT);
}